// LocationMemoryBank_5549097747008
// MI455X (gfx1250) — compile-verified
//
#include <hip/hip_runtime.h>
#include <math.h>

// ---------------------------------------------------------------------------
// LocationMemoryBank on gfx1250 (MI455X, wave32)
//   Phase 1: pairwise squared distances via V_WMMA_F32_16X16X4_F32 using the
//            rank-4 identity  d2 = [qx qy |q|^2 1] . [-2cx -2cy 1 |c|^2]^T
//            (one WMMA = 16 queries x 16 banks). A second WMMA with the
//            grid-rounded query detects exact matches (d2r < 0.25 R^2).
//   Phase 2: memory-bound weighted gather of 8 feature rows per query.
// ---------------------------------------------------------------------------

typedef float v2f __attribute__((ext_vector_type(2)));
typedef float v8f __attribute__((ext_vector_type(8)));

#define RR      5.0e-05f
#define S_SLOTS 20
#define D_DIM   512
#define EQ_T    (0.25f * RR * RR)   // rounded-d2 below this => exact match
#define NR2     (9.0f * RR * RR)    // neighbor radius^2 = (3R)^2
#define FINF    3.0e38f

// Branchless sorted insertion into an ascending top-4 (value, index) list.
__device__ __forceinline__ void ins4(float d, int n,
                                     float& a0, float& a1, float& a2, float& a3,
                                     int& i0, int& i1, int& i2, int& i3) {
  bool c = d < a3; a3 = c ? d : a3; i3 = c ? n : i3;
  c = a3 < a2; { float tf = a2; int ti = i2;
    a2 = c ? a3 : a2; i2 = c ? i3 : i2; a3 = c ? tf : a3; i3 = c ? ti : i3; }
  c = a2 < a1; { float tf = a1; int ti = i1;
    a1 = c ? a2 : a1; i1 = c ? i2 : i1; a2 = c ? tf : a2; i2 = c ? ti : i2; }
  c = a1 < a0; { float tf = a0; int ti = i0;
    a0 = c ? a1 : a0; i0 = c ? i1 : i0; a1 = c ? tf : a1; i1 = c ? ti : i1; }
}

// One wave (32 lanes) handles 16 queries; sweeps all M banks in 16-wide tiles.
__global__ __launch_bounds__(32) void lmb_select(
    const float* __restrict__ qgps,    // (B,2)
    const float* __restrict__ coords,  // (M,2)
    int B, int M,
    int*   __restrict__ meta,          // (B,8) feature-row index m*S + s
    float* __restrict__ wts)           // (B,8) weights
{
  __shared__ float sd[16][16][4];
  __shared__ int   si[16][16][4];
  __shared__ int   se[16][16];

  const int lane = threadIdx.x;       // 0..31
  const int half = lane >> 4;         // 0: K=0,1 lanes  1: K=2,3 lanes
  const int l16  = lane & 15;
  const int qbase = blockIdx.x * 16;

  int qq = qbase + l16; if (qq >= B) qq = B - 1;
  const float qx = qgps[2 * qq + 0];
  const float qy = qgps[2 * qq + 1];
  const float rx = rintf(qx / RR) * RR;
  const float ry = rintf(qy / RR) * RR;

  // A operand (16x4 f32): lanes 0-15 supply K=0,1 = (qx,qy); lanes 16-31
  // supply K=2,3 = (|q|^2, 1).
  v2f a_e, a_r;
  a_e.x = half ? (qx * qx + qy * qy) : qx;
  a_e.y = half ? 1.0f               : qy;
  a_r.x = half ? (rx * rx + ry * ry) : rx;
  a_r.y = half ? 1.0f                : ry;

  // Per-lane state: rows i -> query row (i + 8*half), column = l16.
  float td[8][4];
  int   ti[8][4];
  int   em[8];
#pragma unroll
  for (int i = 0; i < 8; ++i) {
    em[i] = 0x7fffffff;
#pragma unroll
    for (int k = 0; k < 4; ++k) { td[i][k] = FINF; ti[i][k] = 0; }
  }

  const v8f z = {0.f, 0.f, 0.f, 0.f, 0.f, 0.f, 0.f, 0.f};
  const int ntiles = (M + 15) >> 4;

  for (int t = 0; t < ntiles; ++t) {
    const int n  = (t << 4) + l16;          // bank column this lane owns
    const int nc = (n < M) ? n : (M - 1);
    float cx = coords[2 * nc + 0];
    float cy = coords[2 * nc + 1];
    if (n >= M) { cx = 1.0e4f; cy = 1.0e4f; }  // pushes d2 huge, kills eq

    // B operand (4x16 f32): lanes 0-15 supply K=0,1 = (-2cx,-2cy); lanes
    // 16-31 supply K=2,3 = (1, |c|^2) for the same column l16.
    v2f bv;
    bv.x = half ? 1.0f              : (-2.0f * cx);
    bv.y = half ? (cx * cx + cy * cy) : (-2.0f * cy);

    v8f d2  = __builtin_amdgcn_wmma_f32_16x16x4_f32(
        false, a_e, false, bv, (short)0, z, false, false);
    v8f d2r = __builtin_amdgcn_wmma_f32_16x16x4_f32(
        false, a_r, false, bv, (short)0, z, false, false);

#pragma unroll
    for (int i = 0; i < 8; ++i) {
      const float d  = d2[i];
      const bool  eq = d2r[i] < EQ_T;
      em[i] = (eq && n < em[i]) ? n : em[i];
      const float dd = eq ? FINF : d;       // exclude exact hits from k-NN
      ins4(dd, n, td[i][0], td[i][1], td[i][2], td[i][3],
                  ti[i][0], ti[i][1], ti[i][2], ti[i][3]);
    }
  }

  // Spill per-lane candidates; 16 lanes x 4 candidates per query row.
#pragma unroll
  for (int i = 0; i < 8; ++i) {
    const int row = i + (half << 3);
#pragma unroll
    for (int k = 0; k < 4; ++k) {
      sd[row][l16][k] = td[i][k];
      si[row][l16][k] = ti[i][k];
    }
    se[row][l16] = em[i];
  }
  __syncthreads();

  // Lane q (q<16) merges 64 candidates of query row q and emits descriptors.
  if (lane < 16) {
    const int q = qbase + lane;
    if (q < B) {
      float b0 = FINF, b1 = FINF, b2 = FINF, b3 = FINF;
      int   m0 = 0, m1 = 0, m2 = 0, m3 = 0;
      int   emin = 0x7fffffff;
      for (int l = 0; l < 16; ++l) {
        const int e = se[lane][l];
        emin = (e < emin) ? e : emin;
#pragma unroll
        for (int k = 0; k < 4; ++k)
          ins4(sd[lane][l][k], si[lane][l][k], b0, b1, b2, b3, m0, m1, m2, m3);
      }

      const bool hit = emin < M;
      if (hit) {
        float e8[8], s8 = 0.0f;
#pragma unroll
        for (int k = 0; k < 8; ++k) { e8[k] = __expf((float)k); s8 += e8[k]; }
        const float inv = 1.0f / s8;
#pragma unroll
        for (int k = 0; k < 8; ++k) {
          meta[q * 8 + k] = emin * S_SLOTS + (S_SLOTS - 8) + k; // slots 12..19
          wts[q * 8 + k]  = e8[k] * inv;
        }
      } else {
        const int  nn[4] = {m0, m1, m2, m3};
        const bool vv[4] = {b0 < NR2, b1 < NR2, b2 < NR2, b3 < NR2};
        float e[8], s = 0.0f;
#pragma unroll
        for (int j = 0; j < 8; ++j) {
          const float ej = vv[j >> 1] ? __expf((float)j) : 0.0f;
          e[j] = ej; s += ej;
        }
        const float inv = (s > 0.0f) ? (1.0f / s) : 0.0f;
#pragma unroll
        for (int j = 0; j < 8; ++j) {
          meta[q * 8 + j] = nn[j >> 1] * S_SLOTS + (S_SLOTS - 2) + (j & 1);
          wts[q * 8 + j]  = e[j] * inv;  // zero weight for invalid neighbors
        }
      }
    }
  }
}

// Memory-bound gather: out[b,:] = sum_j w[b][j] * feat[src[b][j], :]
__global__ __launch_bounds__(128) void lmb_gather(
    const float* __restrict__ feat,   // (M*S, D)
    const int*   __restrict__ meta,   // (B,8)
    const float* __restrict__ wts,    // (B,8)
    float*       __restrict__ out)    // (B,D)
{
  const int b = blockIdx.x;
  const int d = threadIdx.x << 2;     // 128 threads * float4 = 512 dims
  float4 acc = make_float4(0.f, 0.f, 0.f, 0.f);
#pragma unroll
  for (int j = 0; j < 8; ++j) {
    const int   src = meta[b * 8 + j];   // uniform per block -> scalar load
    const float w   = wts[b * 8 + j];
    const float4 f = *(const float4*)(feat + (size_t)src * D_DIM + d);
    acc.x += w * f.x; acc.y += w * f.y; acc.z += w * f.z; acc.w += w * f.w;
  }
  *(float4*)(out + (size_t)b * D_DIM + d) = acc;
}

extern "C" void kernel_launch(void* const* d_in, const int* in_sizes, int n_in,
                              void* d_out, int out_size, void* d_ws, size_t ws_size,
                              hipStream_t stream) {
  const float* qgps   = (const float*)d_in[0];  // (B,2)
  const float* coords = (const float*)d_in[1];  // (M,2)
  const float* feat   = (const float*)d_in[2];  // (M,S,D)
  const int B = in_sizes[0] / 2;
  const int M = in_sizes[1] / 2;

  int*   meta = (int*)d_ws;
  float* wts  = (float*)((char*)d_ws + (size_t)B * 8 * sizeof(int));

  const int nwaves = (B + 15) / 16;
  lmb_select<<<nwaves, 32, 0, stream>>>(qgps, coords, B, M, meta, wts);
  lmb_gather<<<B, 128, 0, stream>>>(feat, meta, wts, (float*)d_out);
}